// CenterLoss_42468636622951
// MI455X (gfx1250) — compile-verified
//
#include <hip/hip_runtime.h>

// CenterLoss on MI455X (gfx1250, wave32).
// Exploits the label mask: only distmat[b, labels[b]] survives, so this is a
// 1 MB gather + per-row squared distance, not a 51 MB GEMM. The row reduction
// is done on the f32 WMMA pipe: D = diff * diff^T via V_WMMA_F32_16X16X4_F32
// (A-layout of diff == B-layout of diff^T, so the same VGPRs feed SRC0 and
// SRC1); the diagonal of D accumulated over K=128 is ||x_b - c_{l_b}||^2.

typedef __attribute__((ext_vector_type(2))) float v2f;
typedef __attribute__((ext_vector_type(8))) float v8f;

#define BATCH       1024
#define NUM_CLASSES 100000
#define FEAT        128

__global__ void __launch_bounds__(32)
centerloss_init_out(float* __restrict__ out) {
    if (threadIdx.x == 0) {
        // (B*C - B) masked entries each clip to 1e-12; divided by B:
        out[0] = (float)((double)(NUM_CLASSES - 1) * 1e-12);
    }
}

__global__ void __launch_bounds__(32)
centerloss_wmma_kernel(const float* __restrict__ x,
                       const int*   __restrict__ labels,
                       const float* __restrict__ centers,
                       float*       __restrict__ out)
{
    const int lane = threadIdx.x;     // 0..31, wave32
    const int r    = lane & 15;       // matrix row this lane serves
    const int half = lane >> 4;       // K-half: lanes 0-15 -> K%4 in {0,1}, 16-31 -> {2,3}
    const int row  = blockIdx.x * 16 + r;       // 64 blocks * 16 rows = 1024

    const int label = labels[row];
    const float* __restrict__ xr = x       + (size_t)row   * FEAT;
    const float* __restrict__ cr = centers + (size_t)label * FEAT;

    v8f acc = {};  // 16x16 f32 C/D tile, 8 VGPRs/lane

    // K = 128 -> 32 chained 16x16x4 f32 WMMAs. Per chunk each lane supplies
    // diff[r][kbase+0..1] (A VGPR0/1 for its lane group), which simultaneously
    // is the correct B-layout of diff^T.
    #pragma unroll
    for (int kc = 0; kc < 32; ++kc) {
        const int kbase = kc * 4 + half * 2;
        v2f xv = *(const v2f*)(xr + kbase);
        v2f cv = *(const v2f*)(cr + kbase);
        v2f d  = xv - cv;
        // (neg_a, A, neg_b, B, c_mod, C, reuse_a, reuse_b)
        acc = __builtin_amdgcn_wmma_f32_16x16x4_f32(
            false, d, false, d, (short)0, acc, false, false);
    }

    // Diagonal of D: C/D layout has (M = v [+8 for lanes>=16], N = lane&15).
    // M==N lives in lanes 0..7 (acc[r]) and lanes 24..31 (acc[r-8]).
    const bool has_diag = (half == 0) ? (r < 8) : (r >= 8);
    float v = 0.0f;
    if (has_diag) {
        const int idx = r & 7;
        float dsq = 0.0f;
        #pragma unroll
        for (int i = 0; i < 8; ++i)
            if (idx == i) dsq = acc[i];
        dsq = fminf(fmaxf(dsq, 1e-12f), 1e12f);   // jnp.clip(d, 1e-12, 1e12)
        v = dsq * (1.0f / (float)BATCH);
    }

    // Wave32 reduction, then one atomic per wave.
    #pragma unroll
    for (int off = 16; off > 0; off >>= 1)
        v += __shfl_down(v, off, 32);
    if (lane == 0)
        atomicAdd(out, v);
}

extern "C" void kernel_launch(void* const* d_in, const int* in_sizes, int n_in,
                              void* d_out, int out_size, void* d_ws, size_t ws_size,
                              hipStream_t stream) {
    (void)in_sizes; (void)n_in; (void)out_size; (void)d_ws; (void)ws_size;
    const float* x       = (const float*)d_in[0];
    const int*   labels  = (const int*)  d_in[1];
    const float* centers = (const float*)d_in[2];
    float*       out     = (float*)d_out;

    centerloss_init_out<<<1, 32, 0, stream>>>(out);
    centerloss_wmma_kernel<<<BATCH / 16, 32, 0, stream>>>(x, labels, centers, out);
}